// ConvAttn3d_6253472383439
// MI455X (gfx1250) — compile-verified
//
#include <hip/hip_runtime.h>

typedef __attribute__((ext_vector_type(2))) float v2f;
typedef __attribute__((ext_vector_type(8))) float v8f;

#define CCH  64
#define DD   64
#define HHH  64
#define WWW  64
#define SP   (DD * HHH * WWW)   // 262144 voxels per (b,c) slice
#define TAPS 27

__global__ __launch_bounds__(256) void convattn3d_fused(
    const float* __restrict__ Q, const float* __restrict__ K,
    const float* __restrict__ V, const float* __restrict__ pe,
    const float* __restrict__ Wo, const float* __restrict__ bo,
    float* __restrict__ out)
{
    __shared__ float pe_s[CCH * TAPS];        // pe_s[c*27 + t]
    __shared__ float sc_s[TAPS][WWW];         // scores -> attn
    __shared__ float x_s[CCH][WWW + 1];       // x[c][w], padded
    __shared__ float w_s[CCH][CCH + 1];       // W_out[o][c], padded

    const int tid = threadIdx.x;
    const int blk = blockIdx.x;               // b*D*H + d*H + h
    const int h   = blk % HHH;
    const int d   = (blk / HHH) % DD;
    const int b   = blk / (HHH * DD);

    // ---- cooperative staging ----
    for (int i = tid; i < CCH * TAPS; i += 256) pe_s[i] = pe[i];
    for (int i = tid; i < TAPS * WWW; i += 256) ((float*)sc_s)[i] = 0.0f;
    for (int i = tid; i < CCH * CCH; i += 256) w_s[i >> 6][i & 63] = Wo[i];
    __syncthreads();

    const int v  = tid & 63;                  // w coordinate
    const int cg = tid >> 6;                  // channel group 0..3
    const int c0 = cg * 16;

    const long bC    = (long)b * CCH * SP;
    const long plane = (long)d * (HHH * WWW) + h * WWW;
    const long qbase = bC + (long)c0 * SP + plane + v;

    float q[16];
    #pragma unroll
    for (int cc = 0; cc < 16; ++cc) q[cc] = Q[qbase + (long)cc * SP];

    // ---- phase 1: scores (partial over 16 channels, LDS atomic reduce) ----
    for (int t = 0; t < TAPS; ++t) {
        const int di = t / 9 - 1, dj = (t / 3) % 3 - 1, dk = t % 3 - 1;
        const int dd = d + di, hh = h + dj, ww = v + dk;
        const bool inb = ((unsigned)dd < DD) & ((unsigned)hh < HHH) & ((unsigned)ww < WWW);
        float s = 0.0f;
        const float* pec = &pe_s[c0 * 27 + t];
        #pragma unroll
        for (int cc = 0; cc < 16; ++cc) s += q[cc] * pec[cc * 27];
        if (inb) {
            const long kb = bC + (long)c0 * SP + (long)dd * (HHH * WWW) + hh * WWW + ww;
            // warm V lines for phase 3 (global_prefetch_b8)
            __builtin_prefetch(&V[kb], 0, 1);
            #pragma unroll
            for (int cc = 0; cc < 16; ++cc) s += q[cc] * K[kb + (long)cc * SP];
        }
        atomicAdd(&sc_s[t][v], s);            // ds_add_f32
    }
    __syncthreads();

    // ---- phase 2: softmax over 27 taps (one thread per voxel) ----
    if (tid < WWW) {
        const float scale = 0.125f;           // 1/sqrt(64)
        float m = -1e30f;
        #pragma unroll
        for (int t = 0; t < TAPS; ++t) m = fmaxf(m, sc_s[t][tid] * scale);
        float sum = 0.0f;
        #pragma unroll
        for (int t = 0; t < TAPS; ++t) sum += __expf(sc_s[t][tid] * scale - m);
        const float inv = 1.0f / sum;
        #pragma unroll
        for (int t = 0; t < TAPS; ++t)
            sc_s[t][tid] = __expf(sc_s[t][tid] * scale - m) * inv;
    }
    __syncthreads();

    // ---- phase 3: attn-weighted V aggregation into LDS ----
    float x[16];
    #pragma unroll
    for (int cc = 0; cc < 16; ++cc) x[cc] = 0.0f;
    for (int t = 0; t < TAPS; ++t) {
        const int di = t / 9 - 1, dj = (t / 3) % 3 - 1, dk = t % 3 - 1;
        const int dd = d + di, hh = h + dj, ww = v + dk;
        if (((unsigned)dd < DD) & ((unsigned)hh < HHH) & ((unsigned)ww < WWW)) {
            const float a = sc_s[t][v];
            const long vb = bC + (long)c0 * SP + (long)dd * (HHH * WWW) + hh * WWW + ww;
            #pragma unroll
            for (int cc = 0; cc < 16; ++cc) x[cc] += a * V[vb + (long)cc * SP];
        }
    }
    #pragma unroll
    for (int cc = 0; cc < 16; ++cc) x_s[c0 + cc][v] = x[cc];
    __syncthreads();

    // ---- phase 4: out = W_out @ x + b via V_WMMA_F32_16X16X4_F32 ----
    const int wave = tid >> 5;                // 0..7
    const int lane = tid & 31;
    const int ln16 = lane & 15;
    const int hi   = lane >> 4;               // 0: lanes 0-15, 1: lanes 16-31
    const long obase = bC + plane;

    #pragma unroll
    for (int rep = 0; rep < 2; ++rep) {
        const int tile = wave + rep * 8;      // 0..15
        const int ob = (tile >> 2) * 16;      // output-channel block
        const int vb = (tile & 3) * 16;       // voxel block

        v8f acc;
        #pragma unroll
        for (int r = 0; r < 8; ++r) acc[r] = bo[ob + r + hi * 8];

        #pragma unroll
        for (int k0 = 0; k0 < CCH; k0 += 4) {
            v2f a, bm;
            // A (16x4 f32): lane L -> M = L%16; low lanes K={0,1}, high lanes K={2,3}
            a.x = w_s[ob + ln16][k0 + hi * 2 + 0];
            a.y = w_s[ob + ln16][k0 + hi * 2 + 1];
            // B (4x16 f32): lane L -> N = L%16; low lanes K={0,1}, high lanes K={2,3}
            bm.x = x_s[k0 + hi * 2 + 0][vb + ln16];
            bm.y = x_s[k0 + hi * 2 + 1][vb + ln16];
            acc = __builtin_amdgcn_wmma_f32_16x16x4_f32(false, a, false, bm,
                                                        (short)0, acc, false, false);
        }

        // D (16x16 f32): VGPR r -> row ob + r + hi*8, col vb + ln16
        #pragma unroll
        for (int r = 0; r < 8; ++r) {
            const int o = ob + r + hi * 8;
            out[obase + (long)o * SP + vb + ln16] = acc[r];
        }
    }
}

extern "C" void kernel_launch(void* const* d_in, const int* in_sizes, int n_in,
                              void* d_out, int out_size, void* d_ws, size_t ws_size,
                              hipStream_t stream) {
    (void)in_sizes; (void)n_in; (void)d_ws; (void)ws_size; (void)out_size;
    const float* Q  = (const float*)d_in[0];
    const float* K  = (const float*)d_in[1];
    const float* V  = (const float*)d_in[2];
    const float* pe = (const float*)d_in[3];
    const float* Wo = (const float*)d_in[4];
    const float* bo = (const float*)d_in[5];
    float* out = (float*)d_out;

    const int B = 2;
    dim3 grid(B * DD * HHH);                  // 8192 blocks, one (b,d,h) row each
    dim3 block(256);                          // 8 waves of 32
    convattn3d_fused<<<grid, block, 0, stream>>>(Q, K, V, pe, Wo, bo, out);
}